// LengthRegulator_65034394796077
// MI455X (gfx1250) — compile-verified
//
#include <hip/hip_runtime.h>
#include <hip/hip_bf16.h>

// Problem constants (match reference)
constexpr int B_      = 32;
constexpr int C_      = 384;
constexpr int T_      = 1024;
constexpr int MAXLEN_ = 8192;

constexpr int THREADS = 128;           // 4 waves (wave32)
constexpr int VPT     = 4;             // frames per thread -> float4 stores
constexpr int PPB     = THREADS * VPT; // 512 frames per block

typedef unsigned int v4u __attribute__((ext_vector_type(4)));
typedef int          v8i __attribute__((ext_vector_type(8)));
typedef float        v4f __attribute__((ext_vector_type(4)));

__global__ __launch_bounds__(THREADS)
void length_regulator_kernel(const float* __restrict__ x,
                             const int*   __restrict__ dur,
                             float*       __restrict__ out)
{
    __shared__ int scum[T_];       // durations, then in-place cumsum
    __shared__ int spart[THREADS]; // per-thread partial sums for scan

    const int b   = blockIdx.y;
    const int tid = threadIdx.x;

    // ---- Stage dur[b, 0:T] (4 KB, contiguous) into LDS via the Tensor Data
    // ---- Mover.  Wave 0 issues the descriptor (inline asm, 2-group form:
    // ---- VADDR2/VADDR3 NULL => tensor up to 2D).  TENSORcnt tracks it.
    if (tid < 32) {
        const unsigned lds_addr = (unsigned)(size_t)(void*)scum;     // low 32 bits = LDS offset
        const unsigned long long ga =
            (unsigned long long)(size_t)(dur + (size_t)b * T_);

        // D# group 0 (CDNA5 ISA)
        v4u g0;
        g0[0] = 1u;                                        // count=1 (valid), user mode
        g0[1] = lds_addr;                                  // lds_addr[31:0]
        g0[2] = (unsigned)(ga & 0xFFFFFFFFu);              // global_addr[31:0]
        g0[3] = (unsigned)((ga >> 32) & 0x1FFFFFFu)        // global_addr[56:32]
              | (2u << 30);                                // type=2 ("image")

        // D# group 1 (§8.4): data_size=4B; 1D tile 1024 x 1
        v8i g1;
        g1[0] = (int)(2u << 16);            // workgroup_mask=0, data_size=2 (4B)
        g1[1] = (int)((unsigned)T_ << 16);  // tensor_dim0[15:0]=1024 (bits 63:48)
        g1[2] = (int)(1u << 16);            // tensor_dim0 hi=0; tensor_dim1[15:0]=1
        g1[3] = (int)((unsigned)T_ << 16);  // tensor_dim1 hi=0; tile_dim0=1024
        g1[4] = 1;                          // tile_dim1=1; tile_dim2=0
        g1[5] = T_;                         // tensor_dim0_stride[31:0]=1024
        g1[6] = (int)((unsigned)T_ << 16);  // stride0 hi=0; tensor_dim1_stride lo16
        g1[7] = 0;                          // tensor_dim1_stride hi

        asm volatile("tensor_load_to_lds %0, %1"
                     :
                     : "s"(g0), "s"(g1)
                     : "memory");
        __builtin_amdgcn_s_wait_tensorcnt(0);   // s_wait_tensorcnt 0
    }
    __syncthreads();

    // ---- In-place inclusive scan of the 1024 durations (8 elems/thread).
    // volatile on the first pass: data was produced by the TDM, not by
    // compiler-visible stores.
    volatile int* vsc = scum;
    int r[8];
    const int base = tid * 8;
    int run = 0;
    #pragma unroll
    for (int i = 0; i < 8; ++i) { run += vsc[base + i]; r[i] = run; }
    spart[tid] = run;
    __syncthreads();
    #pragma unroll
    for (int off = 1; off < THREADS; off <<= 1) {
        int v = (tid >= off) ? spart[tid - off] : 0;
        __syncthreads();
        spart[tid] += v;
        __syncthreads();
    }
    const int excl = (tid > 0) ? spart[tid - 1] : 0;
    #pragma unroll
    for (int i = 0; i < 8; ++i) scum[base + i] = r[i] + excl;
    __syncthreads();

    const int total = scum[T_ - 1];
    const int p0    = blockIdx.x * PPB + tid * VPT;

    // ---- searchsorted(cum, p, side='right') per frame, clamped to T-1
    int  idx[VPT];
    bool ok[VPT];
    #pragma unroll
    for (int v = 0; v < VPT; ++v) {
        const int p = p0 + v;
        int lo = 0, hi = T_;
        while (lo < hi) {
            const int mid = (lo + hi) >> 1;
            if (scum[mid] <= p) lo = mid + 1; else hi = mid;
        }
        idx[v] = (lo < T_) ? lo : (T_ - 1);
        ok[v]  = p < total;
    }

    // ---- Gather x rows (L2-resident) and stream out with NT stores so the
    // ---- 403 MB output stream does not evict x from L2.
    const float* __restrict__ xb = x + (size_t)b * C_ * T_;
    float*       __restrict__ ob = out + (size_t)b * C_ * MAXLEN_ + p0;

    #pragma unroll 2
    for (int c = 0; c < C_; ++c) {
        const float* xr = xb + (size_t)c * T_;
        __builtin_prefetch(xr + (size_t)8 * T_ + idx[0], 0, 0);  // global_prefetch_b8

        const float a0 = xr[idx[0]];
        const float a1 = xr[idx[1]];
        const float a2 = xr[idx[2]];
        const float a3 = xr[idx[3]];
        v4f rr;
        rr.x = ok[0] ? a0 : 0.0f;
        rr.y = ok[1] ? a1 : 0.0f;
        rr.z = ok[2] ? a2 : 0.0f;
        rr.w = ok[3] ? a3 : 0.0f;
        __builtin_nontemporal_store(rr, (v4f*)(ob + (size_t)c * MAXLEN_));
    }
}

extern "C" void kernel_launch(void* const* d_in, const int* in_sizes, int n_in,
                              void* d_out, int out_size, void* d_ws, size_t ws_size,
                              hipStream_t stream) {
    (void)in_sizes; (void)n_in; (void)d_ws; (void)ws_size; (void)out_size;
    const float* x   = (const float*)d_in[0];
    const int*   dur = (const int*)d_in[1];
    // d_in[2] is max_len (== MAXLEN_ at compile time)
    float* out = (float*)d_out;

    dim3 grid(MAXLEN_ / PPB, B_);   // (16, 32) = 512 blocks, 4 waves each
    length_regulator_kernel<<<grid, THREADS, 0, stream>>>(x, dur, out);
}